// MicroExpressionModel_75359496175869
// MI455X (gfx1250) — compile-verified
//
#include <hip/hip_runtime.h>

// ---------------------------------------------------------------------------
// MicroExpression landmark feature extraction (B,T,68,2)f32 -> (B,T,1023)f32
// Bandwidth-bound gather kernel (152 MB traffic, ~0.2 GFLOP): optimize pure
// data movement on MI455X.
//   * grid = (T/TT, B); block = 256 threads = 8 wave32, one wave per frame
//   * stage TT+1=9 frames (t0-1 .. t0+TT-1) global->LDS with
//     global_load_async_to_lds_b128 (ASYNCcnt), s_wait_asynccnt
//   * per wave: anchor/IOD for frame t and t-1 from LDS; lanes stride over
//     51 vertices (5 outs) and 128 edges (6 outs); results written to an LDS
//     output tile (8 x 1023 floats = 2046 x 16B, exactly chunk-aligned)
//   * drain output tile LDS->global with global_store_async_from_lds_b128
//     (fully coalesced aligned b128 streams; stores are 134/152 MB of traffic)
//   * sin(2a)/cos(2a) via exact double-angle identities (no transcendentals)
// ---------------------------------------------------------------------------

#define TT  8            // frames per block (one per wave)
#define TPB 256          // 8 wave32
#define FRAME_F 136      // 68 pts * 2 floats
#define OUT_F   1023     // 51*5 + 128*6 floats per frame
#define NEDGE_HALF 64    // directed edges 64..127 are the reverses of 0..63

// Edge endpoints, already offset by -17 (index into sel[0..50]).
__device__ __constant__ unsigned char ESRC[NEDGE_HALF] = {
    0,1,2,3,              // left brow chain
    5,6,7,8,              // right brow chain
    10,11,12,             // nose bridge chain
    14,15,16,17,          // nose bottom chain
    19,20,21,22,23,       // left eye chain
    25,26,27,28,29,       // right eye chain
    31,32,33,34,35,36,37,38,39,40,41,   // mouth outer chain
    43,44,45,46,47,48,49, // mouth inner chain
    19,25,31,43,          // eye/mouth closures
    4,5,13,               // brow-nose, nose
    0,4,5,9,              // brow-eye
    22,25,                // eye-nosebridge
    19,22,                // left eye - nose bottom
    25,28,                // right eye - nose bottom
    31,37,                // mouth - nose bottom
    31,37                 // mouth outer-inner
};
__device__ __constant__ unsigned char EDST[NEDGE_HALF] = {
    1,2,3,4,
    6,7,8,9,
    11,12,13,
    15,16,17,18,
    20,21,22,23,24,
    26,27,28,29,30,
    32,33,34,35,36,37,38,39,40,41,42,
    44,45,46,47,48,49,50,
    24,30,42,50,
    10,10,16,
    19,22,25,28,
    10,10,
    14,14,
    18,18,
    14,18,
    43,47
};

// anchor = 0.5*(eye_center + mean(nose_bridge)); riod = 1/(|p39-p42|+eps)
__device__ __forceinline__ void frame_stats(const float* __restrict__ f,
                                            float& ax, float& ay, float& riod) {
    float e39x = f[39 * 2], e39y = f[39 * 2 + 1];
    float e42x = f[42 * 2], e42y = f[42 * 2 + 1];
    float ex = 0.5f * (e39x + e42x);
    float ey = 0.5f * (e39y + e42y);
    float nx = 0.25f * (f[27 * 2] + f[28 * 2] + f[29 * 2] + f[30 * 2]);
    float ny = 0.25f * (f[27 * 2 + 1] + f[28 * 2 + 1] + f[29 * 2 + 1] + f[30 * 2 + 1]);
    ax = 0.5f * (ex + nx);
    ay = 0.5f * (ey + ny);
    float dx = e39x - e42x, dy = e39y - e42y;
    riod = 1.0f / (sqrtf(dx * dx + dy * dy) + 1e-8f);
}

__global__ __launch_bounds__(TPB)
void microexpr_feat_kernel(const float* __restrict__ lm,
                           float* __restrict__ out,
                           int T) {
    __shared__ __align__(16) float sm[(TT + 1) * FRAME_F];  // 9*544 B input tile
    __shared__ __align__(16) float obuf[TT * OUT_F];        // 8*4092 B output tile

    const int t0 = blockIdx.x * TT;
    const int b  = blockIdx.y;
    const float* gbase = lm + (size_t)b * (size_t)T * FRAME_F;

    // ---- async stage frames (t0-1 .. t0+TT-1), clamped, as b128 chunks ----
    // Each frame = 544 B = 34 x 16 B chunks. 9 frames -> 306 chunks.
    const int NCH = (TT + 1) * 34;
    for (int i = threadIdx.x; i < NCH; i += TPB) {
        int slot = i / 34;
        int ch   = i - slot * 34;
        int f    = t0 - 1 + slot;
        f = (f < 0) ? 0 : ((f >= T) ? (T - 1) : f);
        unsigned voff = (unsigned)((f * FRAME_F + ch * 4) * 4);   // byte offset
        unsigned la   = (unsigned)(unsigned long long)(&sm[slot * FRAME_F + ch * 4]);
        asm volatile("global_load_async_to_lds_b128 %0, %1, %2"
                     :: "v"(la), "v"(voff), "s"(gbase)
                     : "memory");
    }
    asm volatile("s_wait_asynccnt 0" ::: "memory");
    __syncthreads();

    const int w    = threadIdx.x >> 5;   // wave id == frame within tile
    const int lane = threadIdx.x & 31;
    const int t    = t0 + w;

    if (t < T) {
        const float* cur = &sm[(w + 1) * FRAME_F];
        const float* prv = &sm[w * FRAME_F];
        const bool first = (t == 0);

        float ax, ay, riod, pax, pay, priod;
        frame_stats(cur, ax, ay, riod);
        frame_stats(prv, pax, pay, priod);

        float* ofr = &obuf[w * OUT_F];

        // ---- vertex features: 51 x (dx, dy, dpx, dpy, mag) ----
        for (int v = lane; v < 51; v += 32) {
            const float* p = &cur[(v + 17) * 2];
            const float* q = &prv[(v + 17) * 2];
            float dx = p[0] - ax, dy = p[1] - ay;
            float dpx = first ? 0.0f : (dx - (q[0] - pax));
            float dpy = first ? 0.0f : (dy - (q[1] - pay));
            float mag = sqrtf(dpx * dpx + dpy * dpy) * riod;
            float* o = ofr + v * 5;
            o[0] = dx; o[1] = dy; o[2] = dpx; o[3] = dpy; o[4] = mag;
        }

        // ---- edge features: 128 x (nl, d_nl, sin2, cos2, d_sin, d_cos) ----
        for (int e = lane; e < 128; e += 32) {
            int idx = e & (NEDGE_HALF - 1);
            int s = ESRC[idx], d = EDST[idx];
            if (e >= NEDGE_HALF) { int tmp = s; s = d; d = tmp; }
            int so = (s + 17) * 2, dofs = (d + 17) * 2;

            // current frame
            float dx = cur[so] - cur[dofs];
            float dy = cur[so + 1] - cur[dofs + 1];
            float l2 = dx * dx + dy * dy;
            float nl = sqrtf(l2) * riod;
            float rl2 = (l2 > 0.0f) ? (1.0f / l2) : 0.0f;
            float s2 = 2.0f * dx * dy * rl2;                       // sin(2*atan2)
            float c2 = (l2 > 0.0f) ? ((dx * dx - dy * dy) * rl2)   // cos(2*atan2)
                                   : 1.0f;                         // atan2(0,0)=0

            // previous frame (recompute; VALU is free, we are HBM-bound)
            float px = prv[so] - prv[dofs];
            float py = prv[so + 1] - prv[dofs + 1];
            float pl2 = px * px + py * py;
            float pnl = sqrtf(pl2) * priod;
            float prl2 = (pl2 > 0.0f) ? (1.0f / pl2) : 0.0f;
            float ps2 = 2.0f * px * py * prl2;
            float pc2 = (pl2 > 0.0f) ? ((px * px - py * py) * prl2) : 1.0f;

            float dl  = first ? 0.0f : (nl - pnl);
            float dsn = first ? 0.0f : (s2 - ps2);
            float dcs = first ? 0.0f : (c2 - pc2);

            float* o = ofr + 255 + e * 6;
            o[0] = nl; o[1] = dl; o[2] = s2; o[3] = c2; o[4] = dsn; o[5] = dcs;
        }
    }

    __syncthreads();   // output tile fully resident in LDS

    // ---- drain output tile LDS -> global as coalesced aligned b128 ----
    // Full tile: 8*1023 floats = 2046 exact 16B chunks; block output base is
    // (b*T + t0)*1023*4 bytes = multiple of 32736 -> 16B aligned.
    const int nvalid  = (T - t0 < TT) ? (T - t0) : TT;
    float* obase      = out + ((size_t)b * T + t0) * OUT_F;
    const int total_f = nvalid * OUT_F;
    const int nchunk  = total_f >> 2;
    for (int i = threadIdx.x; i < nchunk; i += TPB) {
        unsigned voff = (unsigned)(i * 16);                         // byte offset
        unsigned la   = (unsigned)(unsigned long long)(&obuf[i * 4]);
        asm volatile("global_store_async_from_lds_b128 %0, %1, %2"
                     :: "v"(voff), "v"(la), "s"(obase)
                     : "memory");
    }
    // tail floats (only when nvalid*1023 % 4 != 0; never for T=128, TT=8)
    const int rem = total_f & 3;
    if ((int)threadIdx.x < rem) {
        int idx = (nchunk << 2) + threadIdx.x;
        obase[idx] = obuf[idx];
    }
    asm volatile("s_wait_asynccnt 0" ::: "memory");  // explicit; S_ENDPGM also waits
}

extern "C" void kernel_launch(void* const* d_in, const int* in_sizes, int n_in,
                              void* d_out, int out_size, void* d_ws, size_t ws_size,
                              hipStream_t stream) {
    (void)n_in; (void)out_size; (void)d_ws; (void)ws_size;
    const float* lm = (const float*)d_in[0];
    float* out = (float*)d_out;
    const int T = 128;                              // reference: (256,128,68,2)
    const int B = in_sizes[0] / (T * FRAME_F);      // = 256
    dim3 grid((T + TT - 1) / TT, B);
    microexpr_feat_kernel<<<grid, TPB, 0, stream>>>(lm, out, T);
}